// MCSoftmaxDenseFA_2284922602284
// MI455X (gfx1250) — compile-verified
//
#include <hip/hip_runtime.h>
#include <math.h>

typedef __attribute__((ext_vector_type(2))) float v2f;
typedef __attribute__((ext_vector_type(8))) float v8f;

#define B_DIM 64
#define D_DIM 2048
#define C_DIM 1000
#define R_DIM 15
#define S_DIM 1000
#define S_PER_BLOCK 25

// ---------------------------------------------------------------------------
// Phase 1: Y[64,N] = X[64,2048] @ W[2048,N] + bias, optional softplus (OP==1).
// One wave computes a 16(M) x 64(N) tile with V_WMMA_F32_16X16X4_F32,
// K consumed 4 at a time. Block = 256 threads = 8 waves -> 64(M) x 128(N).
// N is a template constant so B-pair loads use immediate offsets and all
// column-guarding is hoisted out of the K-loop (clamped addresses; OOB
// columns compute garbage that is never stored).
// ---------------------------------------------------------------------------
template <int N, int OP>
__device__ __forceinline__ void store_tile_f32(v8f acc, const float* __restrict__ bias,
                                               float* __restrict__ Y,
                                               int m0, int nn, int mbase) {
    if (nn >= N) return;
    float bv = bias[nn];
#pragma unroll
    for (int vi = 0; vi < 8; ++vi) {
        float v = acc[vi] + bv;
        if (OP == 1) {                       // softplus + MIN_SCALE
            v = (v > 20.0f) ? v : log1pf(__expf(v));
            v += 0.001f;
        }
        Y[(m0 + mbase + vi) * N + nn] = v;
    }
}

template <int N, int OP>
__global__ __launch_bounds__(256) void gemm_xw_wmma(
    const float* __restrict__ X, const float* __restrict__ W,
    const float* __restrict__ bias, float* __restrict__ Y)
{
    const int lane = threadIdx.x & 31;
    const int wave = threadIdx.x >> 5;
    const int m0   = (wave & 3) * 16;                       // M tile (64 rows / 4 waves)
    const int n0   = blockIdx.x * 128 + (wave >> 2) * 64;   // N tile base (64 cols / wave)

    const int mrow  = lane & 15;
    const int khalf = (lane >> 4) * 2;   // lanes 0-15: K+0/K+1 ; lanes 16-31: K+2/K+3
    const int ncol  = lane & 15;

    v8f acc0 = {}, acc1 = {}, acc2 = {}, acc3 = {};
    const float* __restrict__ arow = X + (m0 + mrow) * D_DIM + khalf;

    // Loop-invariant clamped column addresses: unconditional loads inside the
    // K-loop; out-of-range columns produce garbage only in discarded D columns.
    const int nc0 = min(n0 + ncol,      N - 1);
    const int nc1 = min(n0 + ncol + 16, N - 1);
    const int nc2 = min(n0 + ncol + 32, N - 1);
    const int nc3 = min(n0 + ncol + 48, N - 1);
    const float* __restrict__ bp0 = W + (long)khalf * N + nc0;
    const float* __restrict__ bp1 = W + (long)khalf * N + nc1;
    const float* __restrict__ bp2 = W + (long)khalf * N + nc2;
    const float* __restrict__ bp3 = W + (long)khalf * N + nc3;

#pragma unroll 4
    for (int kk = 0; kk < D_DIM; kk += 4) {
        v2f a;
        a.x = arow[kk];
        a.y = arow[kk + 1];
        const long o = (long)kk * N;        // constant-folded per unrolled body
        v2f b0; b0.x = bp0[o]; b0.y = bp0[o + N];
        v2f b1; b1.x = bp1[o]; b1.y = bp1[o + N];
        v2f b2; b2.x = bp2[o]; b2.y = bp2[o + N];
        v2f b3; b3.x = bp3[o]; b3.y = bp3[o + N];

        acc0 = __builtin_amdgcn_wmma_f32_16x16x4_f32(false, a, false, b0, (short)0, acc0, false, false);
        acc1 = __builtin_amdgcn_wmma_f32_16x16x4_f32(false, a, false, b1, (short)0, acc1, false, false);
        acc2 = __builtin_amdgcn_wmma_f32_16x16x4_f32(false, a, false, b2, (short)0, acc2, false, false);
        acc3 = __builtin_amdgcn_wmma_f32_16x16x4_f32(false, a, false, b3, (short)0, acc3, false, false);
    }

    const int mbase = (lane >> 4) * 8;   // C/D layout: VGPR vi holds M = vi (+8 for lanes 16-31)
    store_tile_f32<N, OP>(acc0, bias, Y, m0, n0 + 0  + ncol, mbase);
    store_tile_f32<N, OP>(acc1, bias, Y, m0, n0 + 16 + ncol, mbase);
    store_tile_f32<N, OP>(acc2, bias, Y, m0, n0 + 32 + ncol, mbase);
    store_tile_f32<N, OP>(acc3, bias, Y, m0, n0 + 48 + ncol, mbase);
}

// ---------------------------------------------------------------------------
// Phase 2: fused einsum + softmax + mean-accumulate.
// grid = (S chunks, B). Each thread owns classes c = tid + j*256 (j<4);
// its 4x15 factor rows live in registers. Per-sample: dot, block softmax,
// accumulate probs into LDS; flush once via atomicAdd.
// ---------------------------------------------------------------------------
__global__ __launch_bounds__(256) void mc_softmax_kernel(
    const float* __restrict__ locs, const float* __restrict__ fl,
    const float* __restrict__ dsc,  const float* __restrict__ diag_eps,
    const float* __restrict__ std_eps, float* __restrict__ acc_g)
{
    __shared__ float sh_acc[C_DIM];
    __shared__ float sh_eps[16];
    __shared__ float sh_red[8];

    const int b    = blockIdx.y;
    const int tid  = threadIdx.x;
    const int lane = tid & 31;
    const int wave = tid >> 5;

    float rfl[4][R_DIM];
    float rloc[4], rdsc[4];
#pragma unroll
    for (int j = 0; j < 4; ++j) {
        int c = tid + j * 256;
        if (c < C_DIM) {
            rloc[j] = locs[b * C_DIM + c];
            rdsc[j] = dsc[b * C_DIM + c];
            const float* fr = fl + ((long)b * C_DIM + c) * R_DIM;
#pragma unroll
            for (int r = 0; r < R_DIM; ++r) rfl[j][r] = fr[r];
        } else {
            rloc[j] = 0.f; rdsc[j] = 0.f;
#pragma unroll
            for (int r = 0; r < R_DIM; ++r) rfl[j][r] = 0.f;
        }
    }
    for (int i = tid; i < C_DIM; i += 256) sh_acc[i] = 0.f;
    __syncthreads();

    const int s_beg = blockIdx.x * S_PER_BLOCK;
    const int s_end = min(s_beg + S_PER_BLOCK, S_DIM);

    for (int s = s_beg; s < s_end; ++s) {
        if (tid < 16) {
            sh_eps[tid] = (tid < R_DIM) ? std_eps[((long)b * S_DIM + s) * R_DIM + tid]
                                        : diag_eps[b * S_DIM + s];
        }
        __syncthreads();

        const float de = sh_eps[15];
        float lat[4];
        float lmax = -1e30f;
#pragma unroll
        for (int j = 0; j < 4; ++j) {
            int c = tid + j * 256;
            float v = -1e30f;
            if (c < C_DIM) {
                v = rloc[j] + rdsc[j] * de;
#pragma unroll
                for (int r = 0; r < R_DIM; ++r) v += rfl[j][r] * sh_eps[r];
            }
            lat[j] = v;
            lmax = fmaxf(lmax, v);
        }
        // block max (wave32 shuffle + cross-wave LDS)
#pragma unroll
        for (int off = 16; off > 0; off >>= 1)
            lmax = fmaxf(lmax, __shfl_xor(lmax, off, 32));
        if (lane == 0) sh_red[wave] = lmax;
        __syncthreads();
        float gmax = sh_red[0];
#pragma unroll
        for (int w = 1; w < 8; ++w) gmax = fmaxf(gmax, sh_red[w]);
        __syncthreads();   // sh_red reused below

        float lsum = 0.f;
#pragma unroll
        for (int j = 0; j < 4; ++j) {
            int c = tid + j * 256;
            if (c < C_DIM) {
                float e = __expf(lat[j] - gmax);
                lat[j] = e;
                lsum += e;
            }
        }
#pragma unroll
        for (int off = 16; off > 0; off >>= 1)
            lsum += __shfl_xor(lsum, off, 32);
        if (lane == 0) sh_red[wave] = lsum;
        __syncthreads();
        float gsum = 0.f;
#pragma unroll
        for (int w = 0; w < 8; ++w) gsum += sh_red[w];
        float inv = 1.0f / gsum;
#pragma unroll
        for (int j = 0; j < 4; ++j) {
            int c = tid + j * 256;
            if (c < C_DIM) sh_acc[c] += lat[j] * inv;   // fixed c<->tid map, no race
        }
        __syncthreads();   // protect sh_eps/sh_red before next sample
    }

    for (int i = tid; i < C_DIM; i += 256)
        atomicAdd(&acc_g[b * C_DIM + i], sh_acc[i]);
}

// ---------------------------------------------------------------------------
__global__ __launch_bounds__(256) void zero_kernel(float* __restrict__ p, int n) {
    int i = blockIdx.x * 256 + threadIdx.x;
    if (i < n) p[i] = 0.f;
}

__global__ __launch_bounds__(256) void finalize_kernel(const float* __restrict__ acc,
                                                       float* __restrict__ out, int n) {
    int i = blockIdx.x * 256 + threadIdx.x;
    if (i < n) {
        float pm = acc[i] * (1.0f / (float)S_DIM);
        pm = fmaxf(pm, 1e-7f);
        out[i] = logf(pm);
    }
}

// ---------------------------------------------------------------------------
extern "C" void kernel_launch(void* const* d_in, const int* in_sizes, int n_in,
                              void* d_out, int out_size, void* d_ws, size_t ws_size,
                              hipStream_t stream) {
    (void)in_sizes; (void)n_in; (void)out_size; (void)ws_size;
    const float* x        = (const float*)d_in[0];
    const float* Wl       = (const float*)d_in[1];
    const float* bl       = (const float*)d_in[2];
    const float* Ws       = (const float*)d_in[3];
    const float* bs       = (const float*)d_in[4];
    const float* Wd       = (const float*)d_in[5];
    const float* bd       = (const float*)d_in[6];
    const float* diag_eps = (const float*)d_in[7];
    const float* std_eps  = (const float*)d_in[8];
    float* out = (float*)d_out;

    float* ws_f   = (float*)d_ws;
    float* w_locs = ws_f;                                   // 64*1000
    float* w_fl   = w_locs + B_DIM * C_DIM;                 // 64*15000
    float* w_dsc  = w_fl   + B_DIM * C_DIM * R_DIM;         // 64*1000
    float* w_acc  = w_dsc  + B_DIM * C_DIM;                 // 64*1000

    dim3 blk(256);
    constexpr int NCR = C_DIM * R_DIM;

    gemm_xw_wmma<C_DIM, 0><<<dim3((C_DIM + 127) / 128), blk, 0, stream>>>(x, Wl, bl, w_locs);
    gemm_xw_wmma<NCR,   0><<<dim3((NCR   + 127) / 128), blk, 0, stream>>>(x, Ws, bs, w_fl);
    gemm_xw_wmma<C_DIM, 1><<<dim3((C_DIM + 127) / 128), blk, 0, stream>>>(x, Wd, bd, w_dsc);

    zero_kernel<<<(B_DIM * C_DIM + 255) / 256, blk, 0, stream>>>(w_acc, B_DIM * C_DIM);

    mc_softmax_kernel<<<dim3((S_DIM + S_PER_BLOCK - 1) / S_PER_BLOCK, B_DIM), blk, 0, stream>>>(
        w_locs, w_fl, w_dsc, diag_eps, std_eps, w_acc);

    finalize_kernel<<<(B_DIM * C_DIM + 255) / 256, blk, 0, stream>>>(w_acc, out, B_DIM * C_DIM);
}